// Householder1x1Conv_6133213298737
// MI455X (gfx1250) — compile-verified
//
#include <hip/hip_runtime.h>
#include <math.h>

// CDNA5 / gfx1250, wave32. Matrix ops via V_WMMA_F32_16X16X4_F32 (exact fp32).

typedef __attribute__((ext_vector_type(2))) float v2f;
typedef __attribute__((ext_vector_type(8))) float v8f;

#define C_DIM   512     // channels (M and K of the GEMM)
#define HW_DIM  4096    // 64*64 spatial columns per batch
#define M_TILE  128
#define N_TILE  128
#define K_TILE  32
#define THREADS 256     // 8 waves of 32

// sB row-pair stride: 2*N_TILE data dwords + 32 pad dwords.
// 288 % 64 == 32  ->  lanes 0-15 (row-pair rp) and lanes 16-31 (rp+1)
// land on disjoint 32-bank halves during fragment reads: conflict-free.
#define SB_STRIDE (2 * N_TILE + 32)

// ---------------------------------------------------------------------------
// Phase 1: W = H_k ... H_1  (sequential Householder chain, 512x512)
// One block, 512 threads; thread j owns column j of W exclusively.
// ---------------------------------------------------------------------------
__global__ __launch_bounds__(C_DIM)
void hh_build_W(const float* __restrict__ V, float* __restrict__ W, int nH) {
    const int j = threadIdx.x;
    __shared__ float sv[C_DIM];
    __shared__ float sred[C_DIM];

    // W = I (column j only -> no cross-thread hazards on W, ever)
    for (int i = 0; i < C_DIM; ++i)
        W[(size_t)i * C_DIM + j] = (i == j) ? 1.0f : 0.0f;

    for (int k = 0; k < nH; ++k) {
        const float vj = V[k * C_DIM + j];
        sv[j]   = vj;
        sred[j] = vj * vj;
        __syncthreads();
        // tree reduction for ||v||^2
        for (int s = C_DIM >> 1; s > 0; s >>= 1) {
            if (j < s) sred[j] += sred[j + s];
            __syncthreads();
        }
        const float inv = 1.0f / (sqrtf(sred[0]) + 1e-8f);   // vhat = sv * inv

        // t[j] = sum_i vhat[i] * W[i][j]   (coalesced across j)
        float t = 0.0f;
        for (int i = 0; i < C_DIM; ++i)
            t += sv[i] * W[(size_t)i * C_DIM + j];
        t *= inv;

        // W[i][j] -= 2 * vhat[i] * t[j]
        const float c2 = 2.0f * inv * t;
        for (int i = 0; i < C_DIM; ++i)
            W[(size_t)i * C_DIM + j] -= c2 * sv[i];
        __syncthreads();   // protect sv/sred reads before next iteration's writes
    }
}

// ---------------------------------------------------------------------------
// Phase 2: Out[b] = W @ X[b]  as fp32 WMMA GEMM.
// Block tile: 128(M) x 128(N), K chunked by 32 through LDS.
// Wave w computes rows [16w, 16w+16) x all 128 N columns (8 accum tiles).
//
// V_WMMA_F32_16X16X4_F32 fragment layout (ISA 7.12.2):
//   A 16x4 : lanes 0-15 = rows M, v0/v1 = K=0/1 ; lanes 16-31 -> K=2/3
//   B 4x16 : lanes 0-15 = cols N, v0/v1 = K=0/1 ; lanes 16-31 -> K=2/3
//   C/D    : VGPR r -> M=r (lanes 0-15) / M=r+8 (lanes 16-31), N=lane%16
//
// LDS layouts are fragment-native: every WMMA source is ONE aligned b64 load
// into an even VGPR pair (no v_mov shuffling).
//   sA[m][k]              : (A[m][kk], A[m][kk+1]) contiguous, stride 36 dwords
//   sB[k>>1][2n + (k&1)]  : (B[kk][n], B[kk+1][n]) contiguous
// ---------------------------------------------------------------------------
__global__ __launch_bounds__(THREADS)
void hh_gemm_wmma(const float* __restrict__ X, const float* __restrict__ W,
                  float* __restrict__ Out) {
    const int tid = threadIdx.x;
    const int bx  = blockIdx.x;            // 0..511 = batch*32 + n-tile
    const int b   = bx >> 5;               // HW_DIM / N_TILE == 32 tiles/batch
    const int n0  = (bx & 31) * N_TILE;
    const int m0  = blockIdx.y * M_TILE;

    const float* xb = X   + (size_t)b * C_DIM * HW_DIM;
    float*       ob = Out + (size_t)b * C_DIM * HW_DIM;

    __shared__ float sA[M_TILE][K_TILE + 4];        // W tile (stride 36 dwords)
    __shared__ float sB[K_TILE / 2][SB_STRIDE];     // X tile, pair-interleaved

    const int wave    = tid >> 5;          // wave32
    const int lane    = tid & 31;
    const int lane_lo = lane & 15;
    const int hi2     = (lane >> 4) << 1;  // 0 for lanes 0-15, 2 for 16-31
    const int m_w     = wave * 16;

    v8f acc[8] = {};

    for (int k0 = 0; k0 < C_DIM; k0 += K_TILE) {
        // ---- stage A: 128x32 floats = 1024 float4, 4 per thread ----
#pragma unroll
        for (int it = 0; it < 4; ++it) {
            const int idx = tid + it * THREADS;          // 0..1023
            const int row = idx >> 3;                    // 8 float4 per row
            const int c4  = (idx & 7) << 2;
            const float4 va = *(const float4*)&W[(size_t)(m0 + row) * C_DIM + k0 + c4];
            *(float4*)&sA[row][c4] = va;
        }
        // ---- stage B: 32x128 floats, pair-interleaved ----
        // unit = (row-pair rp, col-pair cp): 16 * 64 = 1024 units, 4/thread.
        // Write {B[k][n], B[k+1][n], B[k][n+1], B[k+1][n+1]} as one b128:
        // contiguous across the wave -> conflict-free DS stores.
#pragma unroll
        for (int it = 0; it < 4; ++it) {
            const int idx = tid + it * THREADS;          // 0..1023
            const int rp  = idx >> 6;                    // 0..15
            const int cp  = idx & 63;                    // 0..63
            const int k   = rp << 1;
            const int n   = cp << 1;
            const float2 lo = *(const float2*)&xb[(size_t)(k0 + k)     * HW_DIM + n0 + n];
            const float2 hi = *(const float2*)&xb[(size_t)(k0 + k + 1) * HW_DIM + n0 + n];
            float4 wv;
            wv.x = lo.x; wv.y = hi.x; wv.z = lo.y; wv.w = hi.y;
            *(float4*)&sB[rp][n << 1] = wv;
        }
        // pull next chunk toward L2 while we compute (global_prefetch_b8)
        if (k0 + K_TILE < C_DIM) {
            __builtin_prefetch(&W[(size_t)(m0 + (tid & 127)) * C_DIM + k0 + K_TILE], 0, 0);
            __builtin_prefetch(&xb[(size_t)(k0 + K_TILE + (tid >> 3)) * HW_DIM
                                   + n0 + ((tid & 7) << 4)], 0, 0);
        }
        __syncthreads();

        // ---- compute: 8 K-steps of 4, 8 WMMAs each; 9 b64 DS loads / 8 WMMA
#pragma unroll
        for (int ks = 0; ks < K_TILE; ks += 4) {
            const int kk = ks + hi2;                     // even
            const v2f a = *(const v2f*)&sA[m_w + lane_lo][kk];
#pragma unroll
            for (int nt = 0; nt < 8; ++nt) {
                const v2f bf = *(const v2f*)&sB[kk >> 1][(nt * 16 + lane_lo) << 1];
                acc[nt] = __builtin_amdgcn_wmma_f32_16x16x4_f32(
                    /*neg_a=*/false, a, /*neg_b=*/false, bf,
                    /*c_mod=*/(short)0, acc[nt],
                    /*reuse_a=*/false, /*reuse_b=*/false);
            }
        }
        __syncthreads();
    }

    // ---- store: C/D layout -> row = m0 + m_w + r + 8*(lane>=16), col = nt*16 + lane%16
    const int row_base = m0 + m_w + ((lane >> 4) << 3);
#pragma unroll
    for (int nt = 0; nt < 8; ++nt) {
#pragma unroll
        for (int r = 0; r < 8; ++r) {
            ob[(size_t)(row_base + r) * HW_DIM + n0 + nt * 16 + lane_lo] = acc[nt][r];
        }
    }
}

// ---------------------------------------------------------------------------
extern "C" void kernel_launch(void* const* d_in, const int* in_sizes, int n_in,
                              void* d_out, int out_size, void* d_ws, size_t ws_size,
                              hipStream_t stream) {
    const float* x = (const float*)d_in[0];   // [16, 512, 64, 64] fp32
    const float* V = (const float*)d_in[1];   // [32, 512] fp32
    float* out = (float*)d_out;               // [16, 512, 64, 64] fp32
    float* W   = (float*)d_ws;                // 512*512 fp32 = 1 MB scratch

    const int nH = in_sizes[1] / C_DIM;                   // 32
    const int B  = in_sizes[0] / (C_DIM * HW_DIM);        // 16

    hipLaunchKernelGGL(hh_build_W, dim3(1), dim3(C_DIM), 0, stream, V, W, nH);

    dim3 grid(B * (HW_DIM / N_TILE), C_DIM / M_TILE);     // (512, 4)
    hipLaunchKernelGGL(hh_gemm_wmma, grid, dim3(THREADS), 0, stream, x, W, out);
}